// DOLPHIN_DECODER_39307540693948
// MI455X (gfx1250) — compile-verified
//
#include <hip/hip_runtime.h>

// ---------------------------------------------------------------------------
// DOLPHIN decoder prefill for gfx1250 (MI455X), fp32 path via
// V_WMMA_F32_16X16X4_F32. Memory-bound workload (~460 MB @ 23.3 TB/s),
// so each weight element is streamed from HBM exactly once: every GEMM
// block covers the full M=64 with 8 waves (4 M-tiles x 2 N-tiles).
// LDB is a template parameter so all inner-loop addresses are immediate
// offsets (no per-iteration 64-bit multiplies).
// ---------------------------------------------------------------------------

typedef float v2f __attribute__((ext_vector_type(2)));
typedef float v8f __attribute__((ext_vector_type(8)));

#if defined(__HIP_DEVICE_COMPILE__) && !__has_builtin(__builtin_amdgcn_wmma_f32_16x16x4_f32)
#error "gfx1250 f32 WMMA builtin not available on this toolchain"
#endif

constexpr int kS    = 64;    // new tokens
constexpr int kD    = 1024;
constexpr int kH    = 16;
constexpr int kDK   = 64;
constexpr int kDFF  = 4096;
constexpr int kHIST = 448;
constexpr int kKV   = 512;   // HIST + S
constexpr int kENC  = 1024;
constexpr int kL    = 6;

// ---------------------------------------------------------------------------
// Generic batched WMMA GEMM:  C[b] = epi(A[b](64xK) * B[b](KxLDB-led N) + ...)
// block = 256 threads = 8 wave32s: waves laid out 4(M) x 2(N) -> 64x32 tile.
// grid.x = N/32, grid.y = batch.
// EPI: 0 = +bias, 1 = +bias +residual, 2 = relu(+bias), 3 = raw
// ---------------------------------------------------------------------------
template <int EPI, int LDB>
__global__ __launch_bounds__(256) void wmma_gemm(
    const float* __restrict__ A, long sA,
    const float* __restrict__ B, long sB,
    float* __restrict__ C, long sC,
    const float* __restrict__ bias,
    const float* __restrict__ R, long sR,
    int K, int lda, int ldc, int ldr)
{
    A += (long)blockIdx.y * sA;
    B += (long)blockIdx.y * sB;
    C += (long)blockIdx.y * sC;
    if (EPI == 1) R += (long)blockIdx.y * sR;

    const int wave = threadIdx.x >> 5;
    const int lane = threadIdx.x & 31;
    const int hl   = lane >> 4;      // lane half: 0 or 1
    const int l16  = lane & 15;
    const int mt   = (wave & 3) << 4;                        // 0,16,32,48
    const int nt   = (blockIdx.x << 5) + ((wave >> 2) << 4); // 32-wide slice

    // A row pointer for this lane (fragment: lanes 0-15 -> K=0,1; 16-31 -> K=2,3)
    const float* Ap = A + (long)(mt + l16) * lda + 2 * hl;
    // B column pointer, pre-advanced to row 2*hl
    const float* Bp = B + nt + l16 + (long)(2 * hl) * LDB;

    v8f acc = {};

    for (int k0 = 0; k0 < K; k0 += 64) {
        if (k0 + 64 < K) {
            // keep the streamed weight matrix ahead in cache
            __builtin_prefetch(Bp + 64L * LDB, 0, 1);
        }
#pragma unroll
        for (int kk = 0; kk < 64; kk += 4) {
            // A: one b64 load, immediate offset
            const v2f av = *(const v2f*)(Ap + kk);
            // B: two b32 loads, immediate offsets (kk*LDB*4 bytes < 8MB IOFFSET)
            v2f bv;
            bv.x = Bp[(long)kk * LDB];
            bv.y = Bp[(long)kk * LDB + LDB];
#if defined(__HIP_DEVICE_COMPILE__)
            acc = __builtin_amdgcn_wmma_f32_16x16x4_f32(
                false, av, false, bv, (short)0, acc, false, false);
#endif
        }
        Ap += 64;
        Bp += 64L * LDB;
    }

#pragma unroll
    for (int j = 0; j < 8; ++j) {
        // C/D layout: VGPR j holds row mt+j (lanes 0-15) and mt+8+j (16-31)
        const int row = mt + j + 8 * hl;
        const int col = nt + l16;
        float v = acc[j];
        if (EPI == 0 || EPI == 1 || EPI == 2) v += bias[col];
        if (EPI == 1) v += R[(long)row * ldr + col];
        if (EPI == 2) v = fmaxf(v, 0.0f);
        C[(long)row * ldc + col] = v;
    }
}

// ---------------------------------------------------------------------------
// Token embedding + positional encoding:  x[s,d] = emb[id[s],d] + pe[HIST+s,d]
// ---------------------------------------------------------------------------
__global__ void embed_kernel(const int* __restrict__ ids,
                             const float* __restrict__ emb,
                             const float* __restrict__ pe,
                             float* __restrict__ x)
{
    const int s = blockIdx.x;
    const long id = ids[s];
    for (int d = threadIdx.x; d < kD; d += 256)
        x[(long)s * kD + d] = emb[id * kD + d] + pe[(long)(kHIST + s) * kD + d];
}

// ---------------------------------------------------------------------------
// LayerNorm over D=1024, one block per row (256 threads x 4 elements)
// ---------------------------------------------------------------------------
__global__ void layernorm_kernel(const float* __restrict__ in,
                                 const float* __restrict__ g,
                                 const float* __restrict__ b,
                                 float* __restrict__ out)
{
    const int row = blockIdx.x;
    const int tid = threadIdx.x;
    const float* xr = in + (long)row * kD;
    float* orow = out + (long)row * kD;

    __shared__ float red[256];
    float v[4];
    float s = 0.0f;
#pragma unroll
    for (int i = 0; i < 4; ++i) { v[i] = xr[tid + 256 * i]; s += v[i]; }
    red[tid] = s; __syncthreads();
    for (int st = 128; st > 0; st >>= 1) {
        if (tid < st) red[tid] += red[tid + st];
        __syncthreads();
    }
    const float mean = red[0] * (1.0f / kD);
    __syncthreads();

    float vs = 0.0f;
#pragma unroll
    for (int i = 0; i < 4; ++i) { const float d = v[i] - mean; vs += d * d; }
    red[tid] = vs; __syncthreads();
    for (int st = 128; st > 0; st >>= 1) {
        if (tid < st) red[tid] += red[tid + st];
        __syncthreads();
    }
    const float r = rsqrtf(red[0] * (1.0f / kD) + 1e-5f);
#pragma unroll
    for (int i = 0; i < 4; ++i) {
        const int d = tid + 256 * i;
        orow[d] = (v[i] - mean) * r * g[d] + b[d];
    }
}

// ---------------------------------------------------------------------------
// Row softmax with optional causal mask (mask = (j<=s ? 0 : -128)*mask_scale,
// exactly as the reference's sliced-tril mask). grid = (64 rows, 16 heads)
// ---------------------------------------------------------------------------
__global__ void softmax_kernel(float* __restrict__ sc, int kv, int causal,
                               const float* __restrict__ mask_scale)
{
    const int s = blockIdx.x;
    const int h = blockIdx.y;
    float* row = sc + ((long)h * kS + s) * kv;
    const int tid = threadIdx.x;
    const float ms = mask_scale[0];

    __shared__ float red[256];
    float mx = -3.402823466e38f;
    for (int j = tid; j < kv; j += 256) {
        float v = row[j];
        if (causal && j > s) v += -128.0f * ms;
        row[j] = v;
        mx = fmaxf(mx, v);
    }
    red[tid] = mx; __syncthreads();
    for (int st = 128; st > 0; st >>= 1) {
        if (tid < st) red[tid] = fmaxf(red[tid], red[tid + st]);
        __syncthreads();
    }
    mx = red[0];
    __syncthreads();

    float sum = 0.0f;
    for (int j = tid; j < kv; j += 256) {
        const float e = expf(row[j] - mx);
        row[j] = e;
        sum += e;
    }
    red[tid] = sum; __syncthreads();
    for (int st = 128; st > 0; st >>= 1) {
        if (tid < st) red[tid] += red[tid + st];
        __syncthreads();
    }
    const float inv = 1.0f / red[0];
    for (int j = tid; j < kv; j += 256) row[j] *= inv;
}

// ---------------------------------------------------------------------------
// KV-cache assembly
// ---------------------------------------------------------------------------
__global__ void copy_pastk_kernel(const float* __restrict__ src,
                                  float* __restrict__ outk, int n)
{
    const int i = blockIdx.x * 256 + threadIdx.x;
    if (i >= n) return;
    const int t = i % kHIST;            // history step
    const int r = i / kHIST;            // (l*H+h)*DK+dk
    outk[(long)r * kKV + t] = src[i];
}

__global__ void copy_pastv_kernel(const float* __restrict__ src,
                                  float* __restrict__ outv, int n)
{
    const int i = blockIdx.x * 256 + threadIdx.x;
    if (i >= n) return;
    const int g   = i / (kHIST * kDK);  // l*H+h
    const int rem = i % (kHIST * kDK);  // t*DK+dk
    outv[(long)g * kKV * kDK + rem] = src[i];
}

// new k/v [S, H*DK] -> outk_l [H,DK,KV] (transposed) and outv_l [H,KV,DK]
__global__ void scatter_kv_kernel(const float* __restrict__ kb,
                                  const float* __restrict__ vb,
                                  float* __restrict__ outk_l,
                                  float* __restrict__ outv_l)
{
    const int idx = blockIdx.x * 256 + threadIdx.x;  // < S*D
    const int s  = idx >> 10;
    const int c  = idx & 1023;
    const int h  = c >> 6;
    const int dk = c & 63;
    outk_l[((long)(h * kDK + dk)) * kKV + kHIST + s] = kb[idx];
    outv_l[((long)(h * kKV) + kHIST + s) * kDK + dk] = vb[idx];
}

// ---------------------------------------------------------------------------
// Final head: only vocab columns [5,155) are consulted by the argmax.
// ---------------------------------------------------------------------------
__global__ void logits_kernel(const float* __restrict__ xl,
                              const float* __restrict__ w,    // [D, 40000]
                              const float* __restrict__ bias, // [40000]
                              float* __restrict__ lg,
                              int start)
{
    const int c = start + blockIdx.x;
    const int tid = threadIdx.x;
    float acc = 0.0f;
    for (int d = tid; d < kD; d += 256)
        acc += xl[d] * w[(long)d * 40000 + c];
    __shared__ float red[256];
    red[tid] = acc; __syncthreads();
    for (int st = 128; st > 0; st >>= 1) {
        if (tid < st) red[tid] += red[tid + st];
        __syncthreads();
    }
    if (tid == 0) lg[blockIdx.x] = red[0] + bias[c];
}

__global__ void argmax_kernel(const float* __restrict__ lg, int n,
                              float* __restrict__ tail)
{
    const int tid = threadIdx.x;
    __shared__ float rv[256];
    __shared__ int   ri[256];
    float bv = -3.402823466e38f;
    int   bi = 0;
    for (int j = tid; j < n; j += 256) {
        const float v = lg[j];
        if (v > bv) { bv = v; bi = j; }   // strict > keeps first occurrence
    }
    rv[tid] = bv; ri[tid] = bi; __syncthreads();
    for (int st = 128; st > 0; st >>= 1) {
        if (tid < st) {
            if (rv[tid + st] > rv[tid] ||
                (rv[tid + st] == rv[tid] && ri[tid + st] < ri[tid])) {
                rv[tid] = rv[tid + st];
                ri[tid] = ri[tid + st];
            }
        }
        __syncthreads();
    }
    if (tid == 0) {
        tail[0] = (float)ri[0];      // max_ids (relative to language_start)
        tail[1] = (float)kKV;        // kv_len = 512
    }
}

// ---------------------------------------------------------------------------
// Host orchestration
// ---------------------------------------------------------------------------
extern "C" void kernel_launch(void* const* d_in, const int* in_sizes, int n_in,
                              void* d_out, int out_size, void* d_ws, size_t ws_size,
                              hipStream_t stream)
{
    (void)in_sizes; (void)n_in; (void)out_size; (void)ws_size;

    const float* past_k = (const float*)d_in[0];
    const float* past_v = (const float*)d_in[1];
    const float* ck     = (const float*)d_in[2];
    const float* cv     = (const float*)d_in[3];
    const int*   ids    = (const int*)  d_in[4];
    const float* mscale = (const float*)d_in[5];
    const float* emb    = (const float*)d_in[6];
    const float* pe     = (const float*)d_in[7];
    const float* n1g = (const float*)d_in[8],  *n1b = (const float*)d_in[9];
    const float* n2g = (const float*)d_in[10], *n2b = (const float*)d_in[11];
    const float* n3g = (const float*)d_in[12], *n3b = (const float*)d_in[13];
    const float* wq  = (const float*)d_in[14], *bq  = (const float*)d_in[15];
    const float* wk  = (const float*)d_in[16], *bk  = (const float*)d_in[17];
    const float* wv  = (const float*)d_in[18], *bv  = (const float*)d_in[19];
    const float* wo  = (const float*)d_in[20], *bo  = (const float*)d_in[21];
    const float* cwq = (const float*)d_in[22], *cbq = (const float*)d_in[23];
    const float* cwo = (const float*)d_in[24], *cbo = (const float*)d_in[25];
    const float* w1  = (const float*)d_in[26], *b1  = (const float*)d_in[27];
    const float* w2  = (const float*)d_in[28], *b2  = (const float*)d_in[29];
    const float* ag  = (const float*)d_in[30], *ab  = (const float*)d_in[31];
    const float* ow  = (const float*)d_in[32], *ob  = (const float*)d_in[33];

    // outputs: save_k [L,H,DK,512] | save_v [L,H,512,DK] | max_ids | kv_len
    float* outk = (float*)d_out;
    float* outv = outk + (long)kL * kH * kDK * kKV;
    float* tail = outv + (long)kL * kH * kKV * kDK;

    // workspace
    float* x   = (float*)d_ws;
    float* xn  = x   + kS * kD;
    float* qb  = xn  + kS * kD;
    float* kb  = qb  + kS * kD;
    float* vb  = kb  + kS * kD;
    float* ao  = vb  + kS * kD;
    float* sc  = ao  + kS * kD;            // 16*64*1024 worst case
    float* ffh = sc  + (long)kH * kS * kENC;
    float* xl  = ffh + (long)kS * kDFF;
    float* lg  = xl  + kD;

    // embedding + KV history copy (independent of layer loop)
    embed_kernel<<<kS, 256, 0, stream>>>(ids, emb, pe, x);
    {
        const int nk = kL * kH * kDK * kHIST;
        copy_pastk_kernel<<<(nk + 255) / 256, 256, 0, stream>>>(past_k, outk, nk);
        const int nv = kL * kH * kHIST * kDK;
        copy_pastv_kernel<<<(nv + 255) / 256, 256, 0, stream>>>(past_v, outv, nv);
    }

    for (int l = 0; l < kL; ++l) {
        const float* lwq  = wq  + (long)l * kD * kD;
        const float* lwk  = wk  + (long)l * kD * kD;
        const float* lwv  = wv  + (long)l * kD * kD;
        const float* lwo  = wo  + (long)l * kD * kD;
        const float* lcwq = cwq + (long)l * kD * kD;
        const float* lcwo = cwo + (long)l * kD * kD;
        const float* lw1  = w1  + (long)l * kD * kDFF;
        const float* lw2  = w2  + (long)l * kDFF * kD;
        float* okl = outk + (long)l * kH * kDK * kKV;
        float* ovl = outv + (long)l * kH * kKV * kDK;
        const float* ckl = ck + (long)l * kH * kDK * kENC;
        const float* cvl = cv + (long)l * kH * kENC * kDK;

        // ---- self attention ----
        layernorm_kernel<<<kS, 256, 0, stream>>>(x, n1g + (long)l * kD, n1b + (long)l * kD, xn);
        wmma_gemm<0, kD><<<dim3(kD / 32, 1), 256, 0, stream>>>(
            xn, 0, lwq, 0, qb, 0, bq + (long)l * kD, nullptr, 0, kD, kD, kD, 0);
        wmma_gemm<0, kD><<<dim3(kD / 32, 1), 256, 0, stream>>>(
            xn, 0, lwk, 0, kb, 0, bk + (long)l * kD, nullptr, 0, kD, kD, kD, 0);
        wmma_gemm<0, kD><<<dim3(kD / 32, 1), 256, 0, stream>>>(
            xn, 0, lwv, 0, vb, 0, bv + (long)l * kD, nullptr, 0, kD, kD, kD, 0);
        scatter_kv_kernel<<<(kS * kD) / 256, 256, 0, stream>>>(kb, vb, okl, ovl);

        // scores[h] = q_h(64x64) @ K_h(64x512)
        wmma_gemm<3, kKV><<<dim3(kKV / 32, kH), 256, 0, stream>>>(
            qb, 64, okl, (long)kDK * kKV, sc, (long)kS * kKV,
            nullptr, nullptr, 0, kDK, kD, kKV, 0);
        softmax_kernel<<<dim3(kS, kH), 256, 0, stream>>>(sc, kKV, 1, mscale);
        // attn_out[h] = P_h(64x512) @ V_h(512x64)
        wmma_gemm<3, kDK><<<dim3(kDK / 32, kH), 256, 0, stream>>>(
            sc, (long)kS * kKV, ovl, (long)kKV * kDK, ao, 64,
            nullptr, nullptr, 0, kKV, kKV, kD, 0);
        // x = attn_out @ Wo + bo + x   (fused residual)
        wmma_gemm<1, kD><<<dim3(kD / 32, 1), 256, 0, stream>>>(
            ao, 0, lwo, 0, x, 0, bo + (long)l * kD, x, 0, kD, kD, kD, kD);

        // ---- cross attention ----
        layernorm_kernel<<<kS, 256, 0, stream>>>(x, n2g + (long)l * kD, n2b + (long)l * kD, xn);
        wmma_gemm<0, kD><<<dim3(kD / 32, 1), 256, 0, stream>>>(
            xn, 0, lcwq, 0, qb, 0, cbq + (long)l * kD, nullptr, 0, kD, kD, kD, 0);
        wmma_gemm<3, kENC><<<dim3(kENC / 32, kH), 256, 0, stream>>>(
            qb, 64, ckl, (long)kDK * kENC, sc, (long)kS * kENC,
            nullptr, nullptr, 0, kDK, kD, kENC, 0);
        softmax_kernel<<<dim3(kS, kH), 256, 0, stream>>>(sc, kENC, 0, mscale);
        wmma_gemm<3, kDK><<<dim3(kDK / 32, kH), 256, 0, stream>>>(
            sc, (long)kS * kENC, cvl, (long)kENC * kDK, ao, 64,
            nullptr, nullptr, 0, kENC, kENC, kD, 0);
        wmma_gemm<1, kD><<<dim3(kD / 32, 1), 256, 0, stream>>>(
            ao, 0, lcwo, 0, x, 0, cbo + (long)l * kD, x, 0, kD, kD, kD, kD);

        // ---- FFN ----
        layernorm_kernel<<<kS, 256, 0, stream>>>(x, n3g + (long)l * kD, n3b + (long)l * kD, xn);
        wmma_gemm<2, kDFF><<<dim3(kDFF / 32, 1), 256, 0, stream>>>(
            xn, 0, lw1, 0, ffh, 0, b1 + (long)l * kDFF, nullptr, 0,
            kD, kD, kDFF, 0);
        wmma_gemm<1, kD><<<dim3(kD / 32, 1), 256, 0, stream>>>(
            ffh, 0, lw2, 0, x, 0, b2 + (long)l * kD, x, 0,
            kDFF, kDFF, kD, kD);
    }

    // ---- final LN (last token only) + restricted vocab argmax ----
    layernorm_kernel<<<1, 256, 0, stream>>>(x + (long)(kS - 1) * kD, ag, ab, xl);
    logits_kernel<<<150, 256, 0, stream>>>(xl, ow, ob, lg, /*language_start=*/5);
    argmax_kernel<<<1, 256, 0, stream>>>(lg, 150, tail);
}